// Gating_69363721830916
// MI455X (gfx1250) — compile-verified
//
#include <hip/hip_runtime.h>

typedef __attribute__((ext_vector_type(2))) float v2f;
typedef __attribute__((ext_vector_type(8))) float v8f;

#define NTOK   65536     // 8 * 8192 tokens
#define DDIM   1024
#define NEXP   16
#define TOKS_PER_BLOCK 128
#define KTILE  32

// ---------------------------------------------------------------------------
// Pre-kernel: repack W[16][1024] (row-major) into WMMA-B friendly layout:
// WB[k2][e] = float2{ W[e][2*k2], W[e][2*k2+1] },  k2 in [0,512), e in [0,16).
// Per-WMMA B fetch then becomes one contiguous 256B chunk per wave.
// ---------------------------------------------------------------------------
__global__ void repack_w_kernel(const float* __restrict__ W, float* __restrict__ wb) {
    int j  = blockIdx.x * blockDim.x + threadIdx.x;   // 0..8191
    int e  = j & 15;
    int k2 = j >> 4;
    float2 v;
    v.x = W[e * DDIM + 2 * k2];
    v.y = W[e * DDIM + 2 * k2 + 1];
    reinterpret_cast<float2*>(wb)[j] = v;             // coalesced write
}

// ---------------------------------------------------------------------------
// Main gating kernel. Block = 256 threads (8 waves), 128 tokens/block.
// Each wave computes one 16x16 logits tile via V_WMMA_F32_16X16X4_F32.
// ---------------------------------------------------------------------------
__global__ __launch_bounds__(256) void gating_kernel(
    const float* __restrict__ x,    // [65536][1024]
    const float* __restrict__ wb,   // repacked W, [512][16] float2
    const float* __restrict__ bvec, // [16]
    float* __restrict__ out)        // gated [16][65536] then weights [16][65536]
{
    __shared__ __align__(16) float xs[TOKS_PER_BLOCK * 36];   // 32 cols padded->36 (conflict-free, 16B aligned)
    __shared__ float stage[2 * NEXP * 132];                   // padded stride 132 (conflict-free scatter)

    const int tid  = threadIdx.x;
    const int wave = tid >> 5;
    const int l    = tid & 31;
    const int e    = l & 15;        // A: token row select; C/B: expert/column select
    const int half = l >> 4;        // K sub-pair select (A/B), M+8 select (C)
    const int off  = half * 2;
    const int t0   = blockIdx.x * TOKS_PER_BLOCK;

    const float bias = bvec[e];
    const int tlA = wave * 16 + e;  // this lane's token row in the x tile (A matrix)

    v8f acc0 = {};
    v8f acc1 = {};

    for (int kt = 0; kt < DDIM; kt += KTILE) {
        __syncthreads();  // previous tile fully consumed
        // Stage x tile: 128 tokens x 32 cols, float4 coalesced loads.
        for (int i = tid; i < (TOKS_PER_BLOCK * KTILE) / 4; i += 256) {
            int tok = i >> 3;            // 8 float4 per token row
            int grp = i & 7;
            float4 v = reinterpret_cast<const float4*>(x)[(size_t)(t0 + tok) * (DDIM / 4) + (kt >> 2) + grp];
            *reinterpret_cast<float4*>(&xs[tok * 36 + grp * 4]) = v;
        }
        __syncthreads();  // tile ready

        #pragma unroll
        for (int kk = 0; kk < 8; kk += 2) {
            {
                const int d0 = kt + kk * 4;
                v2f a = *reinterpret_cast<const v2f*>(&xs[tlA * 36 + kk * 4 + off]);
                v2f bm = *reinterpret_cast<const v2f*>(&wb[(((d0 >> 1) + half) * 16 + e) * 2]);
                acc0 = __builtin_amdgcn_wmma_f32_16x16x4_f32(
                    false, a, false, bm, (short)0, acc0, false, false);
            }
            {
                const int d0 = kt + (kk + 1) * 4;
                v2f a = *reinterpret_cast<const v2f*>(&xs[tlA * 36 + (kk + 1) * 4 + off]);
                v2f bm = *reinterpret_cast<const v2f*>(&wb[(((d0 >> 1) + half) * 16 + e) * 2]);
                acc1 = __builtin_amdgcn_wmma_f32_16x16x4_f32(
                    false, a, false, bm, (short)0, acc1, false, false);
            }
        }
    }

    // ---- epilogue: bias + softmax over experts + top-2 gate -------------
    // C tile layout: vgpr r, lane l -> token = wave*16 + r + 8*half, expert = l&15.
    float* stageG = stage;
    float* stageW = stage + NEXP * 132;

    #pragma unroll
    for (int r = 0; r < 8; ++r) {
        float logit = acc0[r] + acc1[r] + bias;

        // max over the 16 experts (lanes within each half-wave group)
        float m = logit;
        #pragma unroll
        for (int mask = 1; mask <= 8; mask <<= 1)
            m = fmaxf(m, __shfl_xor(m, mask, 32));
        float ex = __expf(logit - m);
        float s = ex;
        #pragma unroll
        for (int mask = 1; mask <= 8; mask <<= 1)
            s += __shfl_xor(s, mask, 32);
        float w = ex / s;

        // top-1 (argmax, tie -> smaller expert index, matching lax.top_k)
        float v1 = w; int i1 = e;
        #pragma unroll
        for (int mask = 1; mask <= 8; mask <<= 1) {
            float ov = __shfl_xor(v1, mask, 32);
            int   oi = __shfl_xor(i1, mask, 32);
            if (ov > v1 || (ov == v1 && oi < i1)) { v1 = ov; i1 = oi; }
        }
        // top-2 (exclude i1; weights are in (0,1], so -1 acts as -inf)
        float wc = (e == i1) ? -1.0f : w;
        float v2 = wc; int i2 = e;
        #pragma unroll
        for (int mask = 1; mask <= 8; mask <<= 1) {
            float ov = __shfl_xor(v2, mask, 32);
            int   oi = __shfl_xor(i2, mask, 32);
            if (ov > v2 || (ov == v2 && oi < i2)) { v2 = ov; i2 = oi; }
        }
        float g = (e == i1 || e == i2) ? w : 0.0f;

        int tl = wave * 16 + r + half * 8;       // token local index
        stageW[e * 132 + tl] = w;
        stageG[e * 132 + tl] = g;
    }
    __syncthreads();

    // ---- coalesced writeout: [E][tokens] rows of 128 consecutive floats --
    float* outG = out;
    float* outW = out + (size_t)NEXP * NTOK;
    for (int i = tid; i < NEXP * TOKS_PER_BLOCK; i += 256) {
        int ee = i >> 7;
        int c  = i & 127;
        outG[(size_t)ee * NTOK + t0 + c] = stageG[ee * 132 + c];
        outW[(size_t)ee * NTOK + t0 + c] = stageW[ee * 132 + c];
    }
}

extern "C" void kernel_launch(void* const* d_in, const int* in_sizes, int n_in,
                              void* d_out, int out_size, void* d_ws, size_t ws_size,
                              hipStream_t stream) {
    const float* x = (const float*)d_in[0];   // [8,8192,1024]
    const float* W = (const float*)d_in[1];   // [16,1024]
    const float* b = (const float*)d_in[2];   // [16]
    float* out = (float*)d_out;               // gated[16*65536] ++ weights[16*65536]
    float* wb  = (float*)d_ws;                // 64 KB repacked W

    hipLaunchKernelGGL(repack_w_kernel, dim3(32), dim3(256), 0, stream, W, wb);
    hipLaunchKernelGGL(gating_kernel, dim3(NTOK / TOKS_PER_BLOCK), dim3(256), 0, stream,
                       x, wb, b, out);
}